// gconvnet_regression_75505525063922
// MI455X (gfx1250) — compile-verified
//
#include <hip/hip_runtime.h>

typedef float v2f __attribute__((ext_vector_type(2)));
typedef float v8f __attribute__((ext_vector_type(8)));

#define NV 163842
#define NE 983052

struct SelState { unsigned prefix; int kneed; };

__device__ __forceinline__ unsigned fkey(float f) {
    unsigned u = __float_as_uint(f);
    return (u & 0x80000000u) ? ~u : (u | 0x80000000u);
}

// ---------------- utility fills ----------------
__global__ void gcn_fill_f32(float* p, float v, int n) {
    int i = blockIdx.x * blockDim.x + threadIdx.x;
    if (i < n) p[i] = v;
}
__global__ void gcn_fill_i32(int* p, int v, int n) {
    int i = blockIdx.x * blockDim.x + threadIdx.x;
    if (i < n) p[i] = v;
}
__global__ void gcn_fill_u32(unsigned* p, unsigned v, int n) {
    int i = blockIdx.x * blockDim.x + threadIdx.x;
    if (i < n) p[i] = v;
}
__global__ void gcn_seti(int* p, int v) { *p = v; }

// ---------------- edge prep ----------------
__global__ void gcn_edges_i64_to_i32(const long long* __restrict__ ei,
                                     int* __restrict__ s, int* __restrict__ d, int m) {
    int e = blockIdx.x * blockDim.x + threadIdx.x;
    if (e < m) { s[e] = (int)ei[e]; d[e] = (int)ei[(long long)m + e]; }
}

__global__ void gcn_deg(const int* __restrict__ dst, const int* __restrict__ mPtr,
                        float* __restrict__ deg) {
    int m = *mPtr;
    int stride = gridDim.x * blockDim.x;
    for (int e = blockIdx.x * blockDim.x + threadIdx.x; e < m; e += stride)
        atomicAdd(&deg[dst[e]], 1.0f);
}

__global__ void gcn_rsqrt(const float* __restrict__ deg, float* __restrict__ dinv, int n) {
    int i = blockIdx.x * blockDim.x + threadIdx.x;
    if (i < n) dinv[i] = rsqrtf(deg[i]);
}

// ---------------- GEMM: hw = h @ W via f32 WMMA 16x16x4 ----------------
// One wave per 16x16 output tile; W staged in LDS as paired rows (float2) so
// each B fragment is a single ds_load_b64; full row-tiles take an unpredicated
// path with float2 (b64) A loads.
template <int FIN, int FOUT>
__global__ __launch_bounds__(256)
void gcn_gemm_wmma_t(const float* __restrict__ A, const float* __restrict__ W,
                     float* __restrict__ C, int n) {
    constexpr int KT = (FIN < 32) ? FIN : 32;        // K rows staged per LDS tile
    __shared__ float2 ldsW[(KT / 2) * FOUT];         // ldsW[r2*FOUT+c] = (W[2r2][c], W[2r2+1][c])

    int lane = threadIdx.x & 31;
    constexpr int colTiles = FOUT / 16;
    int rowTiles = (n + 15) >> 4;
    int wave = blockIdx.x * 8 + (threadIdx.x >> 5);
    bool active = wave < rowTiles * colTiles;
    int rt = wave / colTiles, ct = wave - rt * colTiles;
    int rowBase = rt << 4, colBase = ct << 4;
    int half = lane >> 4, lrow = lane & 15;
    int arow = rowBase + lrow;
    bool fullTile = active && (rowBase + 16 <= n);
    v8f acc = {};

    for (int k0 = 0; k0 < FIN; k0 += KT) {
        __syncthreads();
        for (int t = threadIdx.x; t < (KT / 2) * FOUT; t += 256) {
            int pr = t / FOUT, col = t - pr * FOUT;
            int r = k0 + pr * 2;
            ldsW[t] = make_float2(W[(size_t)r * FOUT + col],
                                  W[(size_t)(r + 1) * FOUT + col]);
        }
        __syncthreads();
        if (active) {
#pragma unroll
            for (int kk = 0; kk < KT; kk += 4) {
                int ka = kk + half * 2;  // lanes 0-15: K=kk,kk+1 ; lanes 16-31: K=kk+2,kk+3
                v2f a;
                if (fullTile) {
                    a = *(const v2f*)(A + (size_t)arow * FIN + k0 + ka);
                } else {
                    bool ok = arow < n;
                    a.x = ok ? A[(size_t)arow * FIN + k0 + ka] : 0.0f;
                    a.y = ok ? A[(size_t)arow * FIN + k0 + ka + 1] : 0.0f;
                }
                float2 bw = ldsW[(ka >> 1) * FOUT + colBase + lrow];
                v2f b; b.x = bw.x; b.y = bw.y;
                acc = __builtin_amdgcn_wmma_f32_16x16x4_f32(false, a, false, b,
                                                            (short)0, acc, false, false);
            }
        }
    }
    if (!active) return;
    int col = colBase + lrow;
    if (fullTile) {
#pragma unroll
        for (int v = 0; v < 8; ++v)
            C[(size_t)(rowBase + v + half * 8) * FOUT + col] = acc[v];
    } else {
#pragma unroll
        for (int v = 0; v < 8; ++v) {
            int row = rowBase + v + half * 8;
            if (row < n) C[(size_t)row * FOUT + col] = acc[v];
        }
    }
}

// ---------------- aggregation ----------------
__global__ void gcn_init_agg(const float* __restrict__ hw, const float* __restrict__ dinv,
                             float* __restrict__ agg, int total, int F) {
    int idx = blockIdx.x * blockDim.x + threadIdx.x;
    if (idx >= total) return;
    int i = idx / F;
    float di = dinv[i];
    agg[idx] = hw[idx] * di * di;   // self-loop term: hw * (1/deg)
}

__global__ void gcn_scatter(const int* __restrict__ src, const int* __restrict__ dst,
                            const int* __restrict__ mPtr, const float* __restrict__ hw,
                            const float* __restrict__ dinv, float* __restrict__ agg, int F4) {
    int m = *mPtr;
    long long total = (long long)m * F4;
    long long stride = (long long)gridDim.x * blockDim.x;
    for (long long idx = (long long)blockIdx.x * blockDim.x + threadIdx.x;
         idx < total; idx += stride) {
        int e  = (int)(idx / F4);
        int fc = (int)(idx - (long long)e * F4);
        int s = src[e], d = dst[e];
        float nrm = dinv[s] * dinv[d];
        const float4* hp = (const float4*)(hw + (size_t)s * F4 * 4);
        float4 v = hp[fc];
        float* ap = agg + (size_t)d * F4 * 4 + (size_t)fc * 4;
        atomicAdd(ap + 0, v.x * nrm);
        atomicAdd(ap + 1, v.y * nrm);
        atomicAdd(ap + 2, v.z * nrm);
        atomicAdd(ap + 3, v.w * nrm);
    }
}

__global__ void gcn_bias_relu(float* __restrict__ h, const float* __restrict__ b,
                              int total, int F) {
    int idx = blockIdx.x * blockDim.x + threadIdx.x;
    if (idx >= total) return;
    int f = idx % F;
    float v = h[idx] + b[f];
    h[idx] = v > 0.0f ? v : 0.0f;
}

// ---------------- top-k pooling ----------------
__global__ void gcn_pnorm(const float* __restrict__ p, int F, float* __restrict__ pn) {
    __shared__ float sm[256];
    float a = 0.0f;
    for (int f = threadIdx.x; f < F; f += 256) a += p[f] * p[f];
    sm[threadIdx.x] = a; __syncthreads();
    for (int o = 128; o > 0; o >>= 1) {
        if ((int)threadIdx.x < o) sm[threadIdx.x] += sm[threadIdx.x + o];
        __syncthreads();
    }
    if (threadIdx.x == 0) *pn = sqrtf(sm[0]);
}

__global__ void gcn_score(const float* __restrict__ h, const float* __restrict__ p,
                          const float* __restrict__ pn, float* __restrict__ score,
                          int n, int F4) {
    int i = blockIdx.x * blockDim.x + threadIdx.x;
    if (i >= n) return;
    const float4* row = (const float4*)(h + (size_t)i * F4 * 4);
    const float4* pv  = (const float4*)p;
    float acc = 0.0f;
    for (int f = 0; f < F4; ++f) {
        float4 a = row[f], b = pv[f];
        acc += a.x * b.x + a.y * b.y + a.z * b.z + a.w * b.w;
    }
    score[i] = tanhf(acc / *pn);
}

__global__ void gcn_sel_init(SelState* st, int k, int* tieOut) {
    st->prefix = 0; st->kneed = k; tieOut[0] = 0; tieOut[1] = 0;
}

__global__ void gcn_hist(const float* __restrict__ score, int n,
                         unsigned* __restrict__ hist, const SelState* __restrict__ st,
                         int round) {
    int i = blockIdx.x * blockDim.x + threadIdx.x;
    if (i >= n) return;
    unsigned key = fkey(score[i]);
    if (round == 0) {
        atomicAdd(&hist[key >> 21], 1u);
    } else if (round == 1) {
        if ((key >> 21) == st->prefix) atomicAdd(&hist[(key >> 10) & 0x7FFu], 1u);
    } else {
        if ((key >> 10) == st->prefix) atomicAdd(&hist[key & 0x3FFu], 1u);
    }
}

__global__ void gcn_sel_scan(const unsigned* __restrict__ hist, SelState* st, int round) {
    int nbins = (round == 2) ? 1024 : 2048;
    int bits  = (round == 2) ? 10 : 11;
    int kneed = st->kneed;
    long long cum = 0;
    int sel = 0;
    for (int b = nbins - 1; b >= 0; --b) {
        long long c = hist[b];
        if (cum + c >= (long long)kneed) { sel = b; break; }
        cum += c;
    }
    st->kneed = kneed - (int)cum;                       // still needed within bin sel
    st->prefix = (st->prefix << bits) | (unsigned)sel;  // after round 2: full 32-bit key
}

__global__ void gcn_compact(const float* __restrict__ score, int n,
                            const SelState* __restrict__ st, int* __restrict__ newidx,
                            int* __restrict__ perm, int* __restrict__ tieCnt,
                            int* __restrict__ outCnt) {
    int i = blockIdx.x * blockDim.x + threadIdx.x;
    if (i >= n) return;
    unsigned key = fkey(score[i]);
    unsigned T = st->prefix;
    bool keep = false;
    if (key > T) keep = true;
    else if (key == T) { if (atomicAdd(tieCnt, 1) < st->kneed) keep = true; }
    if (keep) {
        int pos = atomicAdd(outCnt, 1);
        perm[pos] = i;
        newidx[i] = pos;
    }
}

__global__ void gcn_pool_copy(const float* __restrict__ h, const float* __restrict__ score,
                              const int* __restrict__ perm, float* __restrict__ hk,
                              int k, int F4) {
    int idx = blockIdx.x * blockDim.x + threadIdx.x;
    if (idx >= k * F4) return;
    int pos = idx / F4, fc = idx - pos * F4;
    int i = perm[pos];
    float s = score[i];
    float4 v = ((const float4*)(h + (size_t)i * F4 * 4))[fc];
    v.x *= s; v.y *= s; v.z *= s; v.w *= s;
    ((float4*)(hk + (size_t)pos * F4 * 4))[fc] = v;
}

__global__ void gcn_remap(const int* __restrict__ s, const int* __restrict__ d,
                          const int* __restrict__ mPtr, const int* __restrict__ newidx,
                          int* __restrict__ so, int* __restrict__ dn,
                          int* __restrict__ mNext) {
    int m = *mPtr;
    int stride = gridDim.x * blockDim.x;
    for (int e = blockIdx.x * blockDim.x + threadIdx.x; e < m; e += stride) {
        int ns = newidx[s[e]], nd = newidx[d[e]];
        if (ns >= 0 && nd >= 0) {
            int pos = atomicAdd(mNext, 1);
            so[pos] = ns; dn[pos] = nd;
        }
    }
}

// ---------------- readout ----------------
__global__ void gcn_reduce_maxmean(const float* __restrict__ h, float* __restrict__ xc,
                                   int n, int F) {
    int f = blockIdx.x;
    float mx = -3.402823466e38f, sm = 0.0f;
    for (int r = threadIdx.x; r < n; r += 256) {
        float v = h[(size_t)r * F + f];
        mx = fmaxf(mx, v); sm += v;
    }
    __shared__ float smx[256], ssm[256];
    smx[threadIdx.x] = mx; ssm[threadIdx.x] = sm; __syncthreads();
    for (int o = 128; o > 0; o >>= 1) {
        if ((int)threadIdx.x < o) {
            smx[threadIdx.x] = fmaxf(smx[threadIdx.x], smx[threadIdx.x + o]);
            ssm[threadIdx.x] += ssm[threadIdx.x + o];
        }
        __syncthreads();
    }
    if (threadIdx.x == 0) { xc[f] = smx[0]; xc[F + f] = ssm[0] / (float)n; }
}

__global__ void gcn_fc(const float* __restrict__ xc, const float* __restrict__ fcw,
                       const float* __restrict__ fcb, const float* __restrict__ fc2w,
                       const float* __restrict__ fc2b, float* __restrict__ out) {
    __shared__ float red[256];
    int j = threadIdx.x;                  // 256 threads
    float acc = fcb[j];
    for (int i = 0; i < 512; ++i) acc += xc[i] * fcw[j * 512 + i];
    float xo = acc > 0.0f ? acc : 0.0f;
    red[j] = xo * fc2w[j]; __syncthreads();
    for (int o = 128; o > 0; o >>= 1) {
        if (j < o) red[j] += red[j + o];
        __syncthreads();
    }
    if (j == 0) out[0] = red[0] + fc2b[0];
}

// ---------------- host driver ----------------
static inline int cdiv(long long a, long long b) { return (int)((a + b - 1) / b); }

extern "C" void kernel_launch(void* const* d_in, const int* in_sizes, int n_in,
                              void* d_out, int out_size, void* d_ws, size_t ws_size,
                              hipStream_t stream) {
    (void)in_sizes; (void)n_in; (void)out_size; (void)ws_size;
    const float*     x   = (const float*)d_in[0];
    const long long* ei  = (const long long*)d_in[1];
    const float* W[4] = {(const float*)d_in[2], (const float*)d_in[5],
                         (const float*)d_in[8], (const float*)d_in[11]};
    const float* B[4] = {(const float*)d_in[3], (const float*)d_in[6],
                         (const float*)d_in[9], (const float*)d_in[12]};
    const float* P[4] = {(const float*)d_in[4], (const float*)d_in[7],
                         (const float*)d_in[10], (const float*)d_in[13]};
    const float* fcw  = (const float*)d_in[14];
    const float* fcb  = (const float*)d_in[15];
    const float* fc2w = (const float*)d_in[16];
    const float* fc2b = (const float*)d_in[17];

    // workspace bump allocator (256B aligned slots)
    char*  base = (char*)d_ws;
    size_t off  = 0;
    auto alloc = [&](size_t bytes) -> void* {
        void* p = base + off;
        off += (bytes + 255) & ~(size_t)255;
        return p;
    };
    const size_t HMAX = 5243136;                 // max n*F over all layers (20481*256)
    float* bufA = (float*)alloc(HMAX * 4);       // pooled features (next layer's input)
    float* bufB = (float*)alloc(HMAX * 4);       // hw = h @ W
    float* bufC = (float*)alloc(HMAX * 4);       // agg / post-ReLU features
    float* deg   = (float*)alloc((size_t)NV * 4);
    float* dinv  = (float*)alloc((size_t)NV * 4);
    float* score = (float*)alloc((size_t)NV * 4);
    int* newidx  = (int*)alloc((size_t)NV * 4);
    int* perm    = (int*)alloc((size_t)NV * 4);
    int* es[2] = {(int*)alloc((size_t)NE * 4), (int*)alloc((size_t)NE * 4)};
    int* ed[2] = {(int*)alloc((size_t)NE * 4), (int*)alloc((size_t)NE * 4)};
    unsigned* hist = (unsigned*)alloc(2048 * 4);
    SelState* st   = (SelState*)alloc(256);
    int* cnt       = (int*)alloc(256);           // [0],[1]=edge counts; [2]=tie; [3]=out
    float* pn      = (float*)alloc(256);
    float* xc      = (float*)alloc(512 * 4);

    const int nl[5]   = {163842, 81921, 40961, 20481, 10241};
    const int Fout[4] = {32, 64, 128, 256};

    // edge list -> int32, edge count[0] = E
    gcn_edges_i64_to_i32<<<cdiv(NE, 256), 256, 0, stream>>>(ei, es[0], ed[0], NE);
    gcn_seti<<<1, 1, 0, stream>>>(cnt + 0, NE);

    int cur = 0;
    for (int l = 0; l < 4; ++l) {
        int n = nl[l], k = nl[l + 1], fo = Fout[l];
        const float* hIn = (l == 0) ? x : bufA;

        // degrees (self loop baked in as 1.0) and D^{-1/2}
        gcn_fill_f32<<<cdiv(n, 256), 256, 0, stream>>>(deg, 1.0f, n);
        gcn_deg<<<1024, 256, 0, stream>>>(ed[cur], cnt + cur, deg);
        gcn_rsqrt<<<cdiv(n, 256), 256, 0, stream>>>(deg, dinv, n);

        // hw = h @ W  (WMMA f32 16x16x4, one wave per 16x16 tile, W staged in LDS)
        int gemmBlocks = cdiv((long long)cdiv(n, 16) * (fo / 16), 8);
        switch (l) {
            case 0: gcn_gemm_wmma_t<4, 32><<<gemmBlocks, 256, 0, stream>>>(hIn, W[l], bufB, n); break;
            case 1: gcn_gemm_wmma_t<32, 64><<<gemmBlocks, 256, 0, stream>>>(hIn, W[l], bufB, n); break;
            case 2: gcn_gemm_wmma_t<64, 128><<<gemmBlocks, 256, 0, stream>>>(hIn, W[l], bufB, n); break;
            case 3: gcn_gemm_wmma_t<128, 256><<<gemmBlocks, 256, 0, stream>>>(hIn, W[l], bufB, n); break;
        }

        // agg = hw/deg (self loop) + scatter of normalized neighbors
        gcn_init_agg<<<cdiv((long long)n * fo, 256), 256, 0, stream>>>(bufB, dinv, bufC, n * fo, fo);
        gcn_scatter<<<4096, 256, 0, stream>>>(es[cur], ed[cur], cnt + cur, bufB, dinv, bufC, fo / 4);
        gcn_bias_relu<<<cdiv((long long)n * fo, 256), 256, 0, stream>>>(bufC, B[l], n * fo, fo);

        // scores
        gcn_pnorm<<<1, 256, 0, stream>>>(P[l], fo, pn);
        gcn_score<<<cdiv(n, 256), 256, 0, stream>>>(bufC, P[l], pn, score, n, fo / 4);

        // exact top-k threshold via 3-round radix select (11/11/10 bits)
        gcn_sel_init<<<1, 1, 0, stream>>>(st, k, cnt + 2);
        for (int r = 0; r < 3; ++r) {
            gcn_fill_u32<<<8, 256, 0, stream>>>(hist, 0u, 2048);
            gcn_hist<<<cdiv(n, 256), 256, 0, stream>>>(score, n, hist, st, r);
            gcn_sel_scan<<<1, 1, 0, stream>>>(hist, st, r);
        }

        // compact kept nodes, gather + scale rows into bufA
        gcn_fill_i32<<<cdiv(n, 256), 256, 0, stream>>>(newidx, -1, n);
        gcn_compact<<<cdiv(n, 256), 256, 0, stream>>>(score, n, st, newidx, perm,
                                                      cnt + 2, cnt + 3);
        gcn_pool_copy<<<cdiv((long long)k * (fo / 4), 256), 256, 0, stream>>>(
            bufC, score, perm, bufA, k, fo / 4);

        // remap surviving edges into ping-pong buffer
        int nxt = cur ^ 1;
        gcn_seti<<<1, 1, 0, stream>>>(cnt + nxt, 0);
        gcn_remap<<<2048, 256, 0, stream>>>(es[cur], ed[cur], cnt + cur, newidx,
                                            es[nxt], ed[nxt], cnt + nxt);
        cur = nxt;
    }

    // readout: per-feature max/mean over 10241 nodes, then FC->ReLU->FC
    gcn_reduce_maxmean<<<256, 256, 0, stream>>>(bufA, xc, nl[4], 256);
    gcn_fc<<<1, 256, 0, stream>>>(xc, fcw, fcb, fc2w, fc2b, (float*)d_out);
}